// AtomicConv_70111046140541
// MI455X (gfx1250) — compile-verified
//
#include <hip/hip_runtime.h>

// AtomicConv-style graph message passing + batchnorm for MI455X (gfx1250).
// N=100000 nodes, E=1.6M edges, K=8 rbf, T=4 feature types.
// Strategy:
//  - feat2 is one-hot in t  -> only 8 atomic f32 adds per contributing edge.
//  - Output (12.8MB) is L2-resident (192MB L2) -> atomics resolve at L2.
//  - BN mean/var reduction runs on the CDNA5 matrix pipe via
//    V_WMMA_F32_16X16X4_F32 with an all-ones B matrix (layout-invariant).
//  - Stats loop trip count is forced scalar (readfirstlane) so EXEC stays
//    all-ones across WMMA (ISA requirement), and is unrolled x2 so both
//    global_load_b64 issue before the s_wait, overlapping memory latency.

typedef __attribute__((ext_vector_type(2))) float v2f;
typedef __attribute__((ext_vector_type(8))) float v8f;

#define NT 4
#define NK 8
#define TKN 32   // NT*NK floats per node

__global__ void zero_kernel(float* __restrict__ out, float* __restrict__ stats,
                            int total) {
  int i = blockIdx.x * blockDim.x + threadIdx.x;
  if (i < total) out[i] = 0.0f;
  if (i < 8) stats[i] = 0.0f;
}

// One thread per edge: compute rp[k] = exp(-g*(d-mu)^2) * 0.5*(cos(pi*d/co)+1)
// (cutoff branch), find the single matching t, scatter 8 fp32 atomics.
__global__ void scatter_kernel(const float* __restrict__ feat,
                               const float* __restrict__ dist,
                               const int* __restrict__ src,
                               const int* __restrict__ dst,
                               const float* __restrict__ cutoffs,
                               const float* __restrict__ means,
                               const float* __restrict__ scal,
                               const float* __restrict__ ftu,
                               float* __restrict__ hv, int E) {
  int e = blockIdx.x * blockDim.x + threadIdx.x;
  if (e >= E) return;
  float d = dist[e];
  int s  = src[e];
  int dd = dst[e];
  float f = feat[s];

  int t = -1;
#pragma unroll
  for (int j = 0; j < NT; ++j)
    if (f == ftu[j]) t = j;
  if (t < 0) return;  // feat2 row is all-zero: edge contributes nothing

  const float kPi = 3.14159265358979323846f;
  float* base = hv + (size_t)dd * TKN + t * NK;
#pragma unroll
  for (int k = 0; k < NK; ++k) {
    float mu = means[k];
    float g  = scal[k];
    float co = cutoffs[k];
    float x  = d - mu;
    float rbf = expf(-g * x * x);
    float cosv = 0.5f * (cosf(kPi * d / co) + 1.0f);
    float cut = (d < co) ? cosv : 0.0f;
    atomicAdd(base + k, f * rbf * cut);
  }
}

// Matrix-pipe reduction: per-t sum and sum-of-squares over hv[N,4,8].
// Per WMMA a wave consumes one node pair (64 floats):
//   A[m, 0..1] = hv[2p  ][m*2 .. m*2+1]   (lanes 0..15, row m = lane)
//   A[m, 2..3] = hv[2p+1][m*2 .. m*2+1]   (lanes 16..31, row m = lane-16)
// Row m covers element offsets m*2, m*2+1 within a node -> t = m/4 is
// constant per row. D = A * ones(4x16) + C accumulates per-row sums,
// replicated across the 16 columns (documented C/D layout).
__global__ void stats_kernel(const float* __restrict__ hv,
                             float* __restrict__ stats, int npairs) {
  int lane = threadIdx.x & 31;
  int half = lane >> 4;      // 0: node 2p (K=0,1)   1: node 2p+1 (K=2,3)
  int row  = lane & 15;      // A-matrix row, t = row/4
  // Force scalar loop control: wave id is wave-uniform; readfirstlane puts it
  // in an SGPR so the loop compiles to s_cmp/s_cbranch and EXEC is untouched
  // across the WMMAs (EXEC must be all-ones for WMMA).
  int wave   = __builtin_amdgcn_readfirstlane(
                   (blockIdx.x * blockDim.x + threadIdx.x) >> 5);
  int nwaves = (gridDim.x * blockDim.x) >> 5;

  const float* base = hv + (size_t)half * TKN + row * 2;  // + p*64 per pair

  v8f cs = {};  // per-row running sums
  v8f cq = {};  // per-row running sums of squares
  v2f ones; ones[0] = 1.0f; ones[1] = 1.0f;

  int p = wave;
  // Unrolled x2: both global_load_b64 issue before first use, so the wave has
  // two loads in flight; prefetch one stride ahead (global_prefetch_b8).
  for (; p + nwaves < npairs; p += 2 * nwaves) {
    const float* p0 = base + (size_t)p * 64;
    const float* p1 = base + (size_t)(p + nwaves) * 64;
    v2f a0; a0[0] = p0[0]; a0[1] = p0[1];
    v2f a1; a1[0] = p1[0]; a1[1] = p1[1];
    __builtin_prefetch(base + (size_t)(p + 2 * nwaves) * 64, 0, 0);
    v2f q0 = a0 * a0;
    v2f q1 = a1 * a1;
    cs = __builtin_amdgcn_wmma_f32_16x16x4_f32(false, a0, false, ones,
                                               (short)0, cs, false, false);
    cq = __builtin_amdgcn_wmma_f32_16x16x4_f32(false, q0, false, ones,
                                               (short)0, cq, false, false);
    cs = __builtin_amdgcn_wmma_f32_16x16x4_f32(false, a1, false, ones,
                                               (short)0, cs, false, false);
    cq = __builtin_amdgcn_wmma_f32_16x16x4_f32(false, q1, false, ones,
                                               (short)0, cq, false, false);
  }
  if (p < npairs) {  // scalar-branch tail (p, npairs uniform)
    const float* p0 = base + (size_t)p * 64;
    v2f a0; a0[0] = p0[0]; a0[1] = p0[1];
    v2f q0 = a0 * a0;
    cs = __builtin_amdgcn_wmma_f32_16x16x4_f32(false, a0, false, ones,
                                               (short)0, cs, false, false);
    cq = __builtin_amdgcn_wmma_f32_16x16x4_f32(false, q0, false, ones,
                                               (short)0, cq, false, false);
  }

  // C/D layout: VGPR r, lanes 0-15 -> row M=r; lanes 16-31 -> row M=r+8.
  // Rows 0-3:t0, 4-7:t1, 8-11:t2, 12-15:t3.
  float sa = cs[0] + cs[1] + cs[2] + cs[3];  // t = 2*half
  float sb = cs[4] + cs[5] + cs[6] + cs[7];  // t = 2*half + 1
  float qa = cq[0] + cq[1] + cq[2] + cq[3];
  float qb = cq[4] + cq[5] + cq[6] + cq[7];

  if (row == 0) {  // values replicated across columns; one lane per half emits
    int t = 2 * half;
    atomicAdd(&stats[t],     sa);
    atomicAdd(&stats[t + 1], sb);
    atomicAdd(&stats[4 + t],     qa);
    atomicAdd(&stats[4 + t + 1], qb);
  }
}

// In-place batchnorm: out = (hv - mean[t]) * rsqrt(var[t]+eps) * w[t] + b[t]
__global__ void bn_kernel(const float* __restrict__ stats,
                          const float* __restrict__ w,
                          const float* __restrict__ b,
                          float* __restrict__ out, int total, float cnt) {
  int i = blockIdx.x * blockDim.x + threadIdx.x;
  if (i >= total) return;
  int t = (i >> 3) & 3;
  float mean = stats[t] / cnt;
  float var  = stats[4 + t] / cnt - mean * mean;
  float inv  = rsqrtf(var + 1e-5f);
  out[i] = (out[i] - mean) * inv * w[t] + b[t];
}

extern "C" void kernel_launch(void* const* d_in, const int* in_sizes, int n_in,
                              void* d_out, int out_size, void* d_ws, size_t ws_size,
                              hipStream_t stream) {
  const float* feat    = (const float*)d_in[0];
  const float* dist    = (const float*)d_in[1];
  const int*   src     = (const int*)d_in[2];
  const int*   dst     = (const int*)d_in[3];
  const float* cutoffs = (const float*)d_in[4];
  const float* means   = (const float*)d_in[5];
  const float* scal    = (const float*)d_in[6];
  const float* ftu     = (const float*)d_in[7];
  const float* bnw     = (const float*)d_in[8];
  const float* bnb     = (const float*)d_in[9];

  int N = in_sizes[0];       // nodes
  int E = in_sizes[1];       // edges
  int K = in_sizes[4];       // rbf count (8)

  float* out   = (float*)d_out;       // accumulates hv, then normalized in place
  float* stats = (float*)d_ws;        // 8 floats: sums[4], sumsq[4]
  int total = N * TKN;                // == out_size

  zero_kernel<<<(total + 255) / 256, 256, 0, stream>>>(out, stats, total);
  scatter_kernel<<<(E + 255) / 256, 256, 0, stream>>>(
      feat, dist, src, dst, cutoffs, means, scal, ftu, out, E);
  stats_kernel<<<1024, 256, 0, stream>>>(out, stats, N / 2);
  bn_kernel<<<(total + 255) / 256, 256, 0, stream>>>(
      stats, bnw, bnb, out, total, (float)N * (float)K);
}